// MultiHeadAttention_51960514347775
// MI455X (gfx1250) — compile-verified
//
#include <hip/hip_runtime.h>
#include <cstdint>

// ---------------------------------------------------------------------------
// MI455X / gfx1250 multi-head attention, fp32 end-to-end using
// V_WMMA_F32_16X16X4_F32 (full-precision f32 matrix pipe, wave32).
// ---------------------------------------------------------------------------

typedef float v2f __attribute__((ext_vector_type(2)));
typedef float v4f __attribute__((ext_vector_type(4)));
typedef float v8f __attribute__((ext_vector_type(8)));

#define D_MODEL 512
#define NHEAD   8
#define DHEAD   64
#define SEQ     2048
#define BATCH   4
#define MROWS   (BATCH * SEQ)   // 8192

__device__ __forceinline__ v8f wmma_f32(v2f a, v2f b, v8f c) {
    // 8 args: (neg_a, A, neg_b, B, c_mod, C, reuse_a, reuse_b)
    return __builtin_amdgcn_wmma_f32_16x16x4_f32(false, a, false, b,
                                                 (short)0, c, false, false);
}

// ---------------------------------------------------------------------------
// C[M,N] = X[M,K] @ W[N,K]^T + bias[N]   (K = N = D_MODEL, fp32 WMMA)
// 256-thread block = 8 waves; the block owns a 128-row x 16-col output panel.
// Wave w computes the 16x16 tile at rows bm + 16w. The 16x64 W tile is loaded
// ONCE per block (8x reuse vs. one-wave blocks); X panel is 128x64 in LDS.
// Rows padded to 68 floats -> conflict-free fragment reads, 16B-aligned.
// mode 0: row-major store C[m][n]
// mode 1: head-split scatter to [B,H,S,dh]:  out[((b*H+h)*S+s)*64 + d]
// ---------------------------------------------------------------------------
__global__ __launch_bounds__(256)
void gemm_xwt_kernel(const float* __restrict__ X,
                     const float* __restrict__ W,
                     const float* __restrict__ bias,
                     float* __restrict__ out,
                     int mode)
{
    __shared__ float lX[128][68];   // 34.8 KB
    __shared__ float lW[16][68];    //  4.3 KB

    const int tid  = threadIdx.x;
    const int lane = tid & 31;
    const int w    = tid >> 5;              // wave 0..7
    const int bm   = blockIdx.y * 128;      // output row panel
    const int bn   = blockIdx.x * 16;       // output col tile

    const int m16 = lane & 15;
    const int ko  = (lane >> 4) * 2;        // A/B fragment K sub-offset

    v8f acc = {};

    for (int k0 = 0; k0 < D_MODEL; k0 += 64) {
        // cooperative coalesced loads: X panel 8192 floats, W tile 1024 floats
        for (int i = 0; i < 8; ++i) {
            int idx = i * 1024 + tid * 4;
            int r = idx >> 6, c = idx & 63;
            *(v4f*)&lX[r][c] =
                *(const v4f*)&X[(size_t)(bm + r) * D_MODEL + k0 + c];
        }
        {
            int idx = tid * 4;
            int r = idx >> 6, c = idx & 63;
            *(v4f*)&lW[r][c] =
                *(const v4f*)&W[(size_t)(bn + r) * D_MODEL + k0 + c];
        }
        __syncthreads();

        for (int kk = 0; kk < 64; kk += 4) {
            v2f a, b;
            // A 16x4 (MxK): lane = m + 16*(k>>1), vgpr = k&1
            a.x = lX[w * 16 + m16][kk + ko];
            a.y = lX[w * 16 + m16][kk + ko + 1];
            // B 4x16 (KxN) = W^T chunk: B[k][n] = W[n][k]
            b.x = lW[m16][kk + ko];
            b.y = lW[m16][kk + ko + 1];
            acc = wmma_f32(a, b, acc);
        }
        __syncthreads();
    }

    // C/D 16x16: vgpr r -> M = r + 8*(lane>>4), N = lane&15
    const int nl = lane & 15;
    const int mh = (lane >> 4) * 8;
    for (int r = 0; r < 8; ++r) {
        int m = bm + w * 16 + mh + r;
        int n = bn + nl;
        float val = acc[r] + bias[n];
        if (mode == 0) {
            out[(size_t)m * D_MODEL + n] = val;
        } else {
            int b = m >> 11, s = m & (SEQ - 1);
            int h = n >> 6,  d = n & (DHEAD - 1);
            out[(((size_t)(b * NHEAD + h) * SEQ + s) << 6) + d] = val;
        }
    }
}

// ---------------------------------------------------------------------------
// Fused attention per (b, h, 16-query tile):
//   scores = (Q Kt)/8 -> mask -> softmax (in LDS) -> attn written to d_out
//   with NON-TEMPORAL stores (512 MB write-once stream must not evict the
//   L2-resident 50 MB QKV working set from the 192 MB L2),
//   ctx = attn @ V  (per-wave WMMA partials combined via ds_add_f32 atomics)
// Q/K/V in [B,H,S,64] layout. 8 waves; each wave owns 16 of the 128 k-tiles.
// LDS: ~138 KB (CDNA5 has 320 KB per WGP).
// ---------------------------------------------------------------------------
__global__ __launch_bounds__(256)
void attn_kernel(const float* __restrict__ Q,
                 const float* __restrict__ K,
                 const float* __restrict__ V,
                 const unsigned char* __restrict__ mask,   // [B,S] bool
                 float* __restrict__ attn_out,             // [B,H,S,S]
                 float* __restrict__ ctx)                  // [B,S,D]
{
    __shared__ float sQ[16][68];          //  4.3 KB
    __shared__ float sS[16][2052];        // 131.3 KB (padded rows)
    __shared__ float sCtx[16][68];        //  4.3 KB
    __shared__ float sRed[16][17];        //  1.1 KB
    __shared__ float sRowMax[16];
    __shared__ float sRowSum[16];

    const int tid  = threadIdx.x;
    const int lane = tid & 31;
    const int w    = tid >> 5;                 // wave id 0..7
    const int qt   = blockIdx.x;               // query tile 0..127
    const int h    = blockIdx.y;
    const int b    = blockIdx.z;
    const size_t head_base = (size_t)(b * NHEAD + h) * SEQ;  // row into [BH,S,64]
    const int bq = qt * 16;

    const int m16 = lane & 15;
    const int ko  = (lane >> 4) * 2;

    // ---- load Q tile (16x64, coalesced 16B) and zero sCtx ----
    {
        int idx = tid * 4;
        int r = idx >> 6, c = idx & 63;
        *(v4f*)&sQ[r][c] =
            *(const v4f*)&Q[(head_base + bq + r) * DHEAD + c];
    }
    for (int e = tid; e < 16 * 68; e += 256)
        (&sCtx[0][0])[e] = 0.0f;
    __syncthreads();

    // ---- scores = Q K^T * 0.125, masked, into sS ----
    for (int t = 0; t < 16; ++t) {
        int bs = (t * 8 + w) * 16;             // this wave's k-tile base
        v8f acc = {};
        for (int kk = 0; kk < DHEAD; kk += 4) {
            v2f a, bf;
            a.x = sQ[m16][kk + ko];
            a.y = sQ[m16][kk + ko + 1];
            // B[k=d][n=s] = K[bs+n][d] : L2-resident gather (50MB QKV << 192MB L2)
            const float* krow = &K[(head_base + bs + m16) * DHEAD + kk + ko];
            bf.x = krow[0];
            bf.y = krow[1];
            acc = wmma_f32(a, bf, acc);
        }
        const int mh = (lane >> 4) * 8;
        for (int r = 0; r < 8; ++r) {
            int m  = mh + r;
            int sk = bs + m16;
            float v = acc[r] * 0.125f;         // 1/sqrt(64)
            sS[m][sk] = mask[b * SEQ + sk] ? -__builtin_inff() : v;
        }
    }
    __syncthreads();

    // ---- softmax over each of the 16 rows (16 threads per row) ----
    {
        int row = tid >> 4;
        int j   = tid & 15;
        float mx = -__builtin_inff();
        for (int c = j; c < SEQ; c += 16) mx = fmaxf(mx, sS[row][c]);
        sRed[row][j] = mx;
        __syncthreads();
        if (j == 0) {
            float m2 = sRed[row][0];
            for (int q = 1; q < 16; ++q) m2 = fmaxf(m2, sRed[row][q]);
            sRowMax[row] = m2;
        }
        __syncthreads();
        float rmax = sRowMax[row];
        float sum = 0.0f;
        for (int c = j; c < SEQ; c += 16) {
            float e = __expf(sS[row][c] - rmax);
            sS[row][c] = e;
            sum += e;
        }
        sRed[row][j] = sum;
        __syncthreads();
        if (j == 0) {
            float s2 = 0.0f;
            for (int q = 0; q < 16; ++q) s2 += sRed[row][q];
            sRowSum[row] = s2;
        }
        __syncthreads();
        float rinv = 1.0f / sRowSum[row];
        for (int c = j; c < SEQ; c += 16) sS[row][c] *= rinv;
    }
    __syncthreads();

    // ---- coalesced NON-TEMPORAL 16B writeback of attn tile [16,2048] ----
    {
        float* aout = attn_out + (head_base + bq) * SEQ;
        for (int i = 0; i < 32; ++i) {
            int flat = (i * 256 + tid) * 4;    // 0..32764
            int r = flat >> 11;
            int c = flat & (SEQ - 1);
            v4f vv = *(v4f*)&sS[r][c];
            __builtin_nontemporal_store(vv, (v4f*)&aout[(size_t)r * SEQ + c]);
        }
    }

    // ---- ctx = attn @ V : per-wave partial over its 16 k-tiles ----
    v8f cacc[4] = {};
    for (int t = 0; t < 16; ++t) {
        int bs = (t * 8 + w) * 16;
        for (int s0 = 0; s0 < 16; s0 += 4) {
            v2f a;
            a.x = sS[m16][bs + s0 + ko];
            a.y = sS[m16][bs + s0 + ko + 1];
            for (int nt = 0; nt < 4; ++nt) {
                v2f bb;
                // B[k=s][n=d] = V[bs+s0+k][nt*16+n] : coalesced L2 hit
                const float* vr0 =
                    &V[(head_base + bs + s0 + ko) * DHEAD + nt * 16 + m16];
                bb.x = vr0[0];
                bb.y = vr0[DHEAD];             // next s row
                cacc[nt] = wmma_f32(a, bb, cacc[nt]);
            }
        }
    }
    // combine the 8 wave partials with LDS float atomics (ds_add_f32)
    {
        const int mh = (lane >> 4) * 8;
        for (int nt = 0; nt < 4; ++nt)
            for (int r = 0; r < 8; ++r)
                atomicAdd(&sCtx[mh + r][nt * 16 + m16], cacc[nt][r]);
    }
    __syncthreads();

    // ---- write ctx tile to [B,S,D] layout (16B, coalesced) ----
    {
        int e = tid * 4;                       // 0..1020 over 16x64
        int m = e >> 6, c = e & 63;
        v4f s = *(v4f*)&sCtx[m][c];
        *(v4f*)&ctx[(size_t)(b * SEQ + bq + m) * D_MODEL + h * DHEAD + c] = s;
    }
}

// ---------------------------------------------------------------------------
extern "C" void kernel_launch(void* const* d_in, const int* in_sizes, int n_in,
                              void* d_out, int out_size, void* d_ws, size_t ws_size,
                              hipStream_t stream)
{
    (void)in_sizes; (void)n_in; (void)out_size; (void)ws_size;

    const float*         query = (const float*)d_in[0];
    const float*         key   = (const float*)d_in[1];
    const float*         value = (const float*)d_in[2];
    const unsigned char* mask  = (const unsigned char*)d_in[3];  // bool [B,S]
    const float* wq = (const float*)d_in[4];
    const float* bq = (const float*)d_in[5];
    const float* wk = (const float*)d_in[6];
    const float* bk = (const float*)d_in[7];
    const float* wv = (const float*)d_in[8];
    const float* bv = (const float*)d_in[9];
    const float* wo = (const float*)d_in[10];
    const float* bo = (const float*)d_in[11];

    float* out  = (float*)d_out;                                  // [B,S,D]
    float* attn = out + (size_t)BATCH * SEQ * D_MODEL;            // [B,H,S,S]

    const size_t TSZ = (size_t)BATCH * SEQ * D_MODEL;             // 4.19M floats
    float* q_ws = (float*)d_ws;          // [B,H,S,64]
    float* k_ws = q_ws + TSZ;
    float* v_ws = k_ws + TSZ;
    float* ctx  = v_ws + TSZ;            // [B,S,D]   (total 64 MB of d_ws)

    dim3 gblk(256);
    dim3 ggrd(D_MODEL / 16, MROWS / 128); // (32, 64) panels

    gemm_xwt_kernel<<<ggrd, gblk, 0, stream>>>(query, wq, bq, q_ws, 1);
    gemm_xwt_kernel<<<ggrd, gblk, 0, stream>>>(key,   wk, bk, k_ws, 1);
    gemm_xwt_kernel<<<ggrd, gblk, 0, stream>>>(value, wv, bv, v_ws, 1);

    attn_kernel<<<dim3(SEQ / 16, NHEAD, BATCH), 256, 0, stream>>>(
        q_ws, k_ws, v_ws, mask, attn, ctx);

    gemm_xwt_kernel<<<ggrd, gblk, 0, stream>>>(ctx, wo, bo, out, 0);
}